// _GATSepModule_17042430231189
// MI455X (gfx1250) — compile-verified
//
#include <hip/hip_runtime.h>
#include <hip/hip_bf16.h>
#include <math.h>

typedef __attribute__((ext_vector_type(2))) float v2f;
typedef __attribute__((ext_vector_type(8))) float v8f;

#define DIM   128
#define HEADS 8
#define HID   512
#define FF_IN 256

// ---------------------------------------------------------------------------
// fp32 WMMA: D = A(16x4) * B(4x16) + C(16x16), full fp32 precision.
// A frag (ISA 7.12.2, 32-bit A 16x4): lane m=lane&15 holds row m;
//   v0,v1 = K = kbase + 2*(lane>=16) .. +1
// B frag (mirrored): lane holds col n=lane&15; v0,v1 = same K pair.
// C/D: VGPR r -> M = r + 8*(lane>=16), N = lane&15.
// ---------------------------------------------------------------------------
static __device__ __forceinline__ v8f wmma4(v2f a, v2f b, v8f c) {
  return __builtin_amdgcn_wmma_f32_16x16x4_f32(false, a, false, b,
                                               (short)0, c, false, false);
}

// monotonic float->uint key so unsigned atomicMax == float max
static __device__ __forceinline__ unsigned fkey(float f) {
  unsigned u = __float_as_uint(f);
  return (u & 0x80000000u) ? ~u : (u | 0x80000000u);
}
static __device__ __forceinline__ float funkey(unsigned k) {
  return __uint_as_float((k & 0x80000000u) ? (k & 0x7FFFFFFFu) : ~k);
}

// ---------------------------------------------------------------------------
// Kernel 1: h1 = h @ W_in + b_in   (fp32 WMMA, wave per 16-row tile)
// ---------------------------------------------------------------------------
__global__ void k_node_proj(const float* __restrict__ h,
                            const float* __restrict__ W,     // [128][128] row-major
                            const float* __restrict__ bias,  // [128]
                            float* __restrict__ h1, int N) {
  const int lane = threadIdx.x & 31;
  const int wave = threadIdx.x >> 5;
  const int row0 = (blockIdx.x * 8 + wave) * 16;
  if (row0 >= N) return;                       // wave-uniform exit

  const int m     = lane & 15;
  const int khalf = (lane >> 4) << 1;          // 0 or 2
  int arow = row0 + m; if (arow >= N) arow = N - 1;  // clamp (stores guarded)
  const float* aptr = h + (size_t)arow * DIM;

  v8f acc[8] = {};
  for (int kt = 0; kt < DIM / 4; ++kt) {
    const int kk = kt * 4 + khalf;
    v2f a = *(const v2f*)(aptr + kk);
    const float* wk = W + (size_t)kk * DIM;
#pragma unroll
    for (int nt = 0; nt < 8; ++nt) {
      const int n = nt * 16 + m;
      v2f b; b.x = wk[n]; b.y = wk[DIM + n];
      acc[nt] = wmma4(a, b, acc[nt]);
    }
  }
  const int rbase = row0 + ((lane >> 4) << 3);
  if (row0 + 16 <= N) {
    // full tile: branchless stores (common case; N=50000 -> 3125/3125 tiles)
    float* op = h1 + (size_t)rbase * DIM + m;
#pragma unroll
    for (int nt = 0; nt < 8; ++nt) {
      const float bv = bias[nt * 16 + m];
#pragma unroll
      for (int r = 0; r < 8; ++r)
        op[(size_t)r * DIM + nt * 16] = acc[nt][r] + bv;
    }
  } else {
#pragma unroll
    for (int nt = 0; nt < 8; ++nt) {
      const int col = nt * 16 + m;
      const float bv = bias[col];
#pragma unroll
      for (int r = 0; r < 8; ++r) {
        const int rr = rbase + r;
        if (rr < N) h1[(size_t)rr * DIM + col] = acc[nt][r] + bv;
      }
    }
  }
}

// ---------------------------------------------------------------------------
// Kernel 2: attn_u = h1@Wu + bu, attn_v = h1@Wv   (tiny GEMMs, VALU)
// block = 256 threads handles 16 nodes; thread -> (row, head, u/v)
// ---------------------------------------------------------------------------
__global__ void k_attn(const float* __restrict__ h1,
                       const float* __restrict__ Wu, const float* __restrict__ bu,
                       const float* __restrict__ Wv,
                       float* __restrict__ au, float* __restrict__ av, int N) {
  __shared__ float rows[16][DIM];
  const int node0 = blockIdx.x * 16;
  for (int i = threadIdx.x; i < 16 * (DIM / 4); i += blockDim.x) {
    const int r = i >> 5, c4 = (i & 31) << 2;
    const int node = node0 + r;
    float4 v = make_float4(0.f, 0.f, 0.f, 0.f);
    if (node < N) v = *(const float4*)(h1 + (size_t)node * DIM + c4);
    *(float4*)&rows[r][c4] = v;
  }
  __syncthreads();
  const int slot  = threadIdx.x & 15;
  const int r     = threadIdx.x >> 4;
  const int head  = slot & 7;
  const int which = slot >> 3;
  const int node  = node0 + r;
  if (node >= N) return;
  const float* Wm = which ? Wv : Wu;
  float s = 0.f;
#pragma unroll 8
  for (int k = 0; k < DIM; ++k) s += rows[r][k] * Wm[k * HEADS + head];
  if (which) av[(size_t)node * HEADS + head] = s;
  else       au[(size_t)node * HEADS + head] = s + bu[head];
}

// ---------------------------------------------------------------------------
// Kernel 3: per-edge scores + LeakyReLU + segment_max (encoded atomicMax)
// ---------------------------------------------------------------------------
__global__ void k_scores(const float* __restrict__ au, const float* __restrict__ av,
                         const int* __restrict__ src, const int* __restrict__ dst,
                         float* __restrict__ sc, unsigned* __restrict__ smax, int E) {
  const int e = blockIdx.x * blockDim.x + threadIdx.x;
  if (e >= E) return;
  const int s = src[e], d = dst[e];
  const float4* up = (const float4*)(au + (size_t)s * HEADS);
  const float4* vp = (const float4*)(av + (size_t)d * HEADS);
  const float4 u0 = up[0], u1 = up[1];
  const float4 v0 = vp[0], v1 = vp[1];
  float v[HEADS] = { u0.x + v0.x, u0.y + v0.y, u0.z + v0.z, u0.w + v0.w,
                     u1.x + v1.x, u1.y + v1.y, u1.z + v1.z, u1.w + v1.w };
#pragma unroll
  for (int hh = 0; hh < HEADS; ++hh)
    v[hh] = (v[hh] >= 0.f) ? v[hh] : 0.2f * v[hh];
  float4* sp = (float4*)(sc + (size_t)e * HEADS);
  sp[0] = make_float4(v[0], v[1], v[2], v[3]);
  sp[1] = make_float4(v[4], v[5], v[6], v[7]);
  unsigned* mp = smax + (size_t)d * HEADS;
#pragma unroll
  for (int hh = 0; hh < HEADS; ++hh) atomicMax(&mp[hh], fkey(v[hh]));
}

// ---------------------------------------------------------------------------
// Kernel 4: e = exp(score - smax[dst]); denom = segment_sum(e)
// ---------------------------------------------------------------------------
__global__ void k_expsum(const int* __restrict__ dst, const unsigned* __restrict__ smax,
                         float* __restrict__ sc, float* __restrict__ denom, int E) {
  const int e = blockIdx.x * blockDim.x + threadIdx.x;
  if (e >= E) return;
  const int d = dst[e];
  const uint4* mp = (const uint4*)(smax + (size_t)d * HEADS);
  const uint4 k0 = mp[0], k1 = mp[1];
  unsigned ks[HEADS] = { k0.x, k0.y, k0.z, k0.w, k1.x, k1.y, k1.z, k1.w };
  float4* sp = (float4*)(sc + (size_t)e * HEADS);
  const float4 s0 = sp[0], s1 = sp[1];
  float s[HEADS] = { s0.x, s0.y, s0.z, s0.w, s1.x, s1.y, s1.z, s1.w };
  float ev[HEADS];
#pragma unroll
  for (int hh = 0; hh < HEADS; ++hh) {
    float mx = funkey(ks[hh]);
    if (!(fabsf(mx) < 3.0e38f)) mx = 0.f;          // NaN/Inf -> 0 (matches ref)
    ev[hh] = expf(s[hh] - mx);
  }
  sp[0] = make_float4(ev[0], ev[1], ev[2], ev[3]);
  sp[1] = make_float4(ev[4], ev[5], ev[6], ev[7]);
  float* dp = denom + (size_t)d * HEADS;
#pragma unroll
  for (int hh = 0; hh < HEADS; ++hh) atomicAdd(&dp[hh], ev[hh]);
}

// ---------------------------------------------------------------------------
// Kernel 5: msg[dst] += h1[src] * prob   (one wave per edge; lane head = lane%8)
// h_heads layout: element d of a row belongs to head d%8.
// ---------------------------------------------------------------------------
__global__ void k_scatter(const int* __restrict__ src, const int* __restrict__ dst,
                          const float* __restrict__ sc, const float* __restrict__ denom,
                          const float* __restrict__ h1, float* __restrict__ msg, int E) {
  const int lane = threadIdx.x & 31;
  const int e = blockIdx.x * (blockDim.x >> 5) + (threadIdx.x >> 5);
  if (e >= E) return;
  const int s = src[e], d = dst[e];
  const int head = lane & 7;                      // (lane+32j)%8 == lane%8
  const float p = sc[(size_t)e * HEADS + head] / denom[(size_t)d * HEADS + head];
  const float* hs = h1 + (size_t)s * DIM;
  float* md = msg + (size_t)d * DIM;
#pragma unroll
  for (int j = 0; j < 4; ++j) {
    const int c = lane + 32 * j;
    atomicAdd(&md[c], hs[c] * p);
  }
}

// ---------------------------------------------------------------------------
// Kernel 6: fused FFN: x = gelu([h1||msg]@W1 + b1); out = x@W2 + b2
// block = 256 (8 waves) per 16-row tile; both GEMMs fp32 WMMA.
// ---------------------------------------------------------------------------
__global__ void k_ffn(const float* __restrict__ h1, const float* __restrict__ msg,
                      const float* __restrict__ W1, const float* __restrict__ b1,
                      const float* __restrict__ W2, const float* __restrict__ b2,
                      float* __restrict__ out, int N) {
  __shared__ float comb[16][FF_IN + 8];   // 16x264  (16.9 KB), rows 16B-aligned
  __shared__ float xs[16][HID + 8];       // 16x520  (33.3 KB)
  const int lane = threadIdx.x & 31;
  const int wave = threadIdx.x >> 5;      // 0..7
  const int row0 = blockIdx.x * 16;

  // stage 0: combined tile -> LDS (vectorized)
  for (int i = threadIdx.x; i < 16 * (FF_IN / 4); i += blockDim.x) {
    const int r = i >> 6, c4 = (i & 63) << 2;
    const int node = row0 + r;
    float4 v = make_float4(0.f, 0.f, 0.f, 0.f);
    if (node < N)
      v = (c4 < DIM) ? *(const float4*)(h1 + (size_t)node * DIM + c4)
                     : *(const float4*)(msg + (size_t)node * DIM + (c4 - DIM));
    *(float4*)&comb[r][c4] = v;
  }
  __syncthreads();

  const int m     = lane & 15;
  const int khalf = (lane >> 4) << 1;
  const int rhalf = (lane >> 4) << 3;

  // stage 1: GEMM1 (K=256) + bias + exact GELU -> xs ; wave owns 64 cols
  {
    v8f acc[4] = {};
    for (int kt = 0; kt < FF_IN / 4; ++kt) {
      const int kk = kt * 4 + khalf;
      v2f a; a.x = comb[m][kk]; a.y = comb[m][kk + 1];
      const float* wk = W1 + (size_t)kk * HID;
#pragma unroll
      for (int nt = 0; nt < 4; ++nt) {
        const int n = wave * 64 + nt * 16 + m;
        v2f b; b.x = wk[n]; b.y = wk[HID + n];
        acc[nt] = wmma4(a, b, acc[nt]);
      }
    }
#pragma unroll
    for (int nt = 0; nt < 4; ++nt) {
      const int col = wave * 64 + nt * 16 + m;
      const float bb = b1[col];
#pragma unroll
      for (int r = 0; r < 8; ++r) {
        const float x = acc[nt][r] + bb;
        xs[r + rhalf][col] = 0.5f * x * (1.f + erff(x * 0.70710678118654752f));
      }
    }
  }
  __syncthreads();

  // stage 2: GEMM2 (K=512) + bias -> out ; wave owns 16 cols
  {
    v8f acc = {};
    for (int kt = 0; kt < HID / 4; ++kt) {
      const int kk = kt * 4 + khalf;
      v2f a; a.x = xs[m][kk]; a.y = xs[m][kk + 1];
      const float* wk = W2 + (size_t)kk * DIM;
      const int n = wave * 16 + m;
      v2f b; b.x = wk[n]; b.y = wk[DIM + n];
      acc = wmma4(a, b, acc);
    }
    const int col = wave * 16 + m;
    const float bb = b2[col];
    if (row0 + 16 <= N) {
      // full tile: branchless stores (always taken for N multiple of 16)
      float* op = out + (size_t)(row0 + rhalf) * DIM + col;
#pragma unroll
      for (int r = 0; r < 8; ++r) op[(size_t)r * DIM] = acc[r] + bb;
    } else {
#pragma unroll
      for (int r = 0; r < 8; ++r) {
        const int rr = row0 + r + rhalf;
        if (rr < N) out[(size_t)rr * DIM + col] = acc[r] + bb;
      }
    }
  }
}

// ---------------------------------------------------------------------------
extern "C" void kernel_launch(void* const* d_in, const int* in_sizes, int n_in,
                              void* d_out, int out_size, void* d_ws, size_t ws_size,
                              hipStream_t stream) {
  const float* h    = (const float*)d_in[0];
  const float* W_in = (const float*)d_in[1];
  const float* b_in = (const float*)d_in[2];
  const float* Wu   = (const float*)d_in[3];
  const float* bu   = (const float*)d_in[4];
  const float* Wv   = (const float*)d_in[5];
  const float* W1   = (const float*)d_in[6];
  const float* b1   = (const float*)d_in[7];
  const float* W2   = (const float*)d_in[8];
  const float* b2   = (const float*)d_in[9];
  const int*   src  = (const int*)d_in[10];
  const int*   dst  = (const int*)d_in[11];
  const int N = in_sizes[0] / DIM;
  const int E = in_sizes[10];
  (void)n_in; (void)out_size; (void)ws_size;

  // workspace carve-out (~107 MB for N=50000, E=1.6M)
  char* ws = (char*)d_ws;
  size_t off = 0;
  float*    h1    = (float*)(ws + off);    off += (size_t)N * DIM * 4;
  float*    au    = (float*)(ws + off);    off += (size_t)N * HEADS * 4;
  float*    av    = (float*)(ws + off);    off += (size_t)N * HEADS * 4;
  unsigned* smax  = (unsigned*)(ws + off); off += (size_t)N * HEADS * 4;
  float*    denom = (float*)(ws + off);    off += (size_t)N * HEADS * 4;
  float*    msg   = (float*)(ws + off);    off += (size_t)N * DIM * 4;
  float*    sc    = (float*)(ws + off);    off += (size_t)E * HEADS * 4;

  hipMemsetAsync(smax,  0, (size_t)N * HEADS * 4, stream);  // 0 == key-identity for max
  hipMemsetAsync(denom, 0, (size_t)N * HEADS * 4, stream);
  hipMemsetAsync(msg,   0, (size_t)N * DIM * 4, stream);

  k_node_proj<<<(N + 127) / 128, 256, 0, stream>>>(h, W_in, b_in, h1, N);
  k_attn     <<<(N + 15) / 16,   256, 0, stream>>>(h1, Wu, bu, Wv, au, av, N);
  k_scores   <<<(E + 255) / 256, 256, 0, stream>>>(au, av, src, dst, sc, smax, E);
  k_expsum   <<<(E + 255) / 256, 256, 0, stream>>>(dst, smax, sc, denom, E);
  k_scatter  <<<(E + 7) / 8,     256, 0, stream>>>(src, dst, sc, denom, h1, msg, E);
  k_ffn      <<<(N + 15) / 16,   256, 0, stream>>>(h1, msg, W1, b1, W2, b2,
                                                   (float*)d_out, N);
}